// BatchContrastLoss_8624294331081
// MI455X (gfx1250) — compile-verified
//
#include <hip/hip_runtime.h>
#include <hip/hip_bf16.h>

typedef __attribute__((ext_vector_type(2))) float v2f;
typedef __attribute__((ext_vector_type(8))) float v8f;
typedef __attribute__((ext_vector_type(4))) unsigned int v4u;
typedef __attribute__((ext_vector_type(8))) int v8i;
typedef __attribute__((ext_vector_type(4))) int v4i;

#define D_DIM 1024
#define NWAVE 8
#define GAMA_INV 14.285714285714286f  // 1/0.07, also the fixed softmax shift C

// ---------------------------------------------------------------------------
// Kernel 1: inverse row norms for Q (rows [0,B)) and Ans (rows [B,2B)).
// inv[row] = 1 / max(||row||_2, 1e-8)
// ---------------------------------------------------------------------------
__global__ __launch_bounds__(128) void rownorm_inv_kernel(
    const float* __restrict__ q, const float* __restrict__ a,
    float* __restrict__ inv, int B, int D) {
  const int row = blockIdx.x;
  const float* src = (row < B) ? (q + (size_t)row * D)
                               : (a + (size_t)(row - B) * D);
  float ss = 0.0f;
  for (int k = threadIdx.x * 4; k < D; k += blockDim.x * 4) {
    const float4 v = *(const float4*)(src + k);
    ss += v.x * v.x + v.y * v.y + v.z * v.z + v.w * v.w;
  }
  __shared__ float red[128];
  red[threadIdx.x] = ss;
  __syncthreads();
  for (int off = 64; off > 0; off >>= 1) {
    if (threadIdx.x < off) red[threadIdx.x] += red[threadIdx.x + off];
    __syncthreads();
  }
  if (threadIdx.x == 0) {
    float n = fmaxf(sqrtf(red[0]), 1e-8f);
    inv[row] = 1.0f / n;
  }
}

// ---------------------------------------------------------------------------
// Kernel 2: fused  Q·A^T  (fp32 WMMA 16x16x4)  +  shifted-softmax loss.
//
// |cos| <= 1  =>  |logit| <= 1/GAMA, so exp(logit - 1/GAMA) never overflows:
// fixed shift replaces the running max, per-lane partial expsums defer the
// 16-lane row reduction to the very end (log-softmax is shift-invariant).
//
// Block: 256 threads (8 waves), owns a 16-row M tile. The 64 KB Q stripe is
// DMA'd into LDS by the Tensor Data Mover (TENSOR_LOAD_TO_LDS, TENSORcnt).
// Each wave processes 2 column tiles per iteration, sharing A-frag DS loads.
// ---------------------------------------------------------------------------
__global__ __launch_bounds__(256) void contrast_loss_kernel(
    const float* __restrict__ q, const float* __restrict__ a,
    const float* __restrict__ qinv, const float* __restrict__ ainv,
    float* __restrict__ out, int B) {
  __shared__ float qs[16 * D_DIM];     // 64 KB Q stripe
  __shared__ float diagv[16];
  __shared__ float sall[NWAVE][16];

  const int i0 = blockIdx.x * 16;
  const int wave = threadIdx.x >> 5;
  const int lane = threadIdx.x & 31;
  const int ln16 = lane & 15;
  const int half = lane >> 4;

#if __has_builtin(__builtin_amdgcn_tensor_load_to_lds)
  // --- TDM: DMA the 16x1024 fp32 Q stripe (64 KB) into LDS. -----------------
  if (wave == 0) {
    const unsigned long long gaddr =
        (unsigned long long)(uintptr_t)(q + (size_t)i0 * D_DIM);
    // Low 32 bits of the generic pointer to a __shared__ object are the LDS
    // byte offset (ISA aperture rule).
    const unsigned int ldsoff = (unsigned int)(uintptr_t)(void*)qs;

    // D# group 0: count=1 | lds_addr | global_addr[56:0] | type=2.
    v4u g0;
    g0[0] = 1u;                                   // count=1, user mode
    g0[1] = ldsoff;                               // lds_addr
    g0[2] = (unsigned int)(gaddr & 0xFFFFFFFFull);  // global_addr[31:0]
    g0[3] = (unsigned int)((gaddr >> 32) & 0x01FFFFFFull) | 0x80000000u;  // type=2

    // D# group 1: 2D tensor 1024x16 (fp32), tile 1024x16, stride 1024.
    v8i g1;
    g1[0] = (int)(2u << 16);                  // workgroup_mask=0, data_size=4B
    g1[1] = (int)((D_DIM & 0xFFFFu) << 16);   // tensor_dim0[15:0] -> bits[63:48]
    g1[2] = (int)((D_DIM >> 16) | (16u << 16));  // tensor_dim0[31:16], tensor_dim1[15:0]
    g1[3] = (int)((D_DIM & 0xFFFFu) << 16);   // tensor_dim1[31:16]=0, tile_dim0=1024
    g1[4] = 16;                               // tile_dim1=16, tile_dim2=0
    g1[5] = D_DIM;                            // tensor_dim0_stride[31:0]
    g1[6] = 0;                                // stride0[47:32]=0, stride1[15:0]=0
    g1[7] = 0;

    const v4i gz4 = {0, 0, 0, 0};             // groups 2/3: unused (2D tensor)
    const v8i gz8 = {0, 0, 0, 0, 0, 0, 0, 0};
    __builtin_amdgcn_tensor_load_to_lds(g0, g1, gz4, gz4, gz8, 0);
    __builtin_amdgcn_s_wait_tensorcnt(0);
  }
#else
  // --- Fallback: coalesced float4 copy of the Q stripe. ---------------------
  {
    const float4* qbase = (const float4*)(q + (size_t)i0 * D_DIM);
    float4* qs4 = (float4*)qs;
    for (int t = threadIdx.x; t < 16 * D_DIM / 4; t += blockDim.x)
      qs4[t] = qbase[t];
  }
#endif
  __syncthreads();

  // Per-row scale (1/||q_i||)/gamma, hoisted: row = i0 + r + 8*half.
  float qrs[8];
#pragma unroll
  for (int r = 0; r < 8; ++r) qrs[r] = qinv[i0 + r + 8 * half] * GAMA_INV;

  // A-fragment source: Q[row = ln16][k + 2*half .. +1] from LDS.
  const float* qrow = qs + ln16 * D_DIM + 2 * half;

  float s[8];
#pragma unroll
  for (int r = 0; r < 8; ++r) s[r] = 0.0f;

  // Per-tile epilogue: scale to logits, accumulate exp(x - C) per lane,
  // capture the diagonal logit when this tile sits on the diagonal.
  auto epilogue = [&](const v8f& c, int j0) {
    const float as = ainv[j0 + ln16];  // per-column 1/||a_j||
#pragma unroll
    for (int r = 0; r < 8; ++r) {
      const float x = c[r] * qrs[r] * as;          // logits_{i,j}
      if (j0 == i0 && ln16 == r + 8 * half) diagv[r + 8 * half] = x;
      s[r] += __expf(x - GAMA_INV);                // shifted exp, no overflow
    }
  };

  const int ntile = B / 16;
  for (int tp = wave * 2; tp < ntile; tp += NWAVE * 2) {
    const int j0a = tp * 16;
    const int j0b = j0a + 16;
    // B-fragment sources: Ans[col = j0+ln16][k + 2*half .. +1] (L2-resident).
    const float* arow0 = a + (size_t)(j0a + ln16) * D_DIM + 2 * half;
    const float* arow1 = arow0 + (size_t)16 * D_DIM;

    // Warm WGP cache for the next tile pair (global_prefetch_b8).
    if (tp + NWAVE * 2 < ntile) {
      const float* pf = arow0 + (size_t)NWAVE * 2 * 16 * D_DIM;
      __builtin_prefetch(pf, 0, 3);
      __builtin_prefetch(pf + (size_t)16 * D_DIM, 0, 3);
    }

    v8f c0 = {}, c1 = {};
#pragma unroll 8
    for (int k = 0; k < D_DIM; k += 4) {
      const v2f av = *(const v2f*)(qrow + k);
      const v2f bv0 = *(const v2f*)(arow0 + k);
      const v2f bv1 = *(const v2f*)(arow1 + k);
      c0 = __builtin_amdgcn_wmma_f32_16x16x4_f32(
          false, av, false, bv0, (short)0, c0, false, false);
      c1 = __builtin_amdgcn_wmma_f32_16x16x4_f32(
          false, av, false, bv1, (short)0, c1, false, false);
    }
    epilogue(c0, j0a);
    epilogue(c1, j0b);
  }

  // One row-wise reduction at the end: sum s[r] over the 16 lanes of a half.
#pragma unroll
  for (int r = 0; r < 8; ++r) {
    float e = s[r];
    e += __shfl_xor(e, 1, 32);
    e += __shfl_xor(e, 2, 32);
    e += __shfl_xor(e, 4, 32);
    e += __shfl_xor(e, 8, 32);
    if (ln16 == r) sall[wave][r + 8 * half] = e;
  }
  __syncthreads();

  // Merge the 8 waves: logsumexp_j = C + log(sum of shifted exps), then
  // loss_i = logsumexp - logits_ii; accumulate mean into out.
  if (threadIdx.x < 16) {
    const int M = threadIdx.x;
    float S = 0.0f;
#pragma unroll
    for (int w = 0; w < NWAVE; ++w) S += sall[w][M];
    const float lossi = (GAMA_INV + logf(S)) - diagv[M];
    atomicAdd(out, lossi / (float)B);
  }
}

// ---------------------------------------------------------------------------
extern "C" void kernel_launch(void* const* d_in, const int* in_sizes, int n_in,
                              void* d_out, int out_size, void* d_ws,
                              size_t ws_size, hipStream_t stream) {
  const float* q = (const float*)d_in[0];
  const float* a = (const float*)d_in[1];
  const int D = D_DIM;
  const int B = in_sizes[0] / D;  // 4096

  float* qinv = (float*)d_ws;
  float* ainv = qinv + B;
  float* out = (float*)d_out;

  (void)hipMemsetAsync(out, 0, sizeof(float) * out_size, stream);
  rownorm_inv_kernel<<<2 * B, 128, 0, stream>>>(q, a, qinv, B, D);
  contrast_loss_kernel<<<B / 16, 256, 0, stream>>>(q, a, qinv, ainv, out, B);
}